// BatchedChebLayer_83064667505061
// MI455X (gfx1250) — compile-verified
//
#include <hip/hip_runtime.h>
#include <hip/hip_bf16.h>

// BatchedChebLayer for MI455X (gfx1250, wave32).
//
// Pipeline (all on `stream`, graph-capture safe):
//   1) init:   tx1 = 0, tx2 = -x                      (fuses the "-tx_0" term)
//   2) spmm:   tx1 += val * x[col]        (atomics, L2-resident: h = 102 MB < 192 MB L2)
//   3) spmm:   tx2 += 2*val * tx1[col]
//   4) gemm:   out = x@W0 + tx1@W1 + tx2@W2 + bias via V_WMMA_F32_16X16X4_F32
//
// Workspace layout: tx1 [T*N*C f32] | tx2 [T*N*C f32]  (2 * 102.4 MB).

typedef float v2f __attribute__((ext_vector_type(2)));
typedef float v4f __attribute__((ext_vector_type(4)));
typedef float v8f __attribute__((ext_vector_type(8)));

#define T_DIM   4
#define N_NODES 50000
#define E_EDGES 800000
#define C_DIM   128
#define K_CHEB  3
#define M_TILES (N_NODES / 16)   // 3125, exact

// ---------------------------------------------------------------------------
// Kernel 1: tx1 = 0 ; tx2 = -x   (float4 vectorized)
// ---------------------------------------------------------------------------
__global__ void cheb_init_kernel(const float* __restrict__ x,
                                 float* __restrict__ tx1,
                                 float* __restrict__ tx2) {
  const long n4 = (long)T_DIM * N_NODES * C_DIM / 4;
  long i = (long)blockIdx.x * blockDim.x + threadIdx.x;
  if (i >= n4) return;
  const v4f xv = ((const v4f*)x)[i];
  v4f z = {0.0f, 0.0f, 0.0f, 0.0f};
  ((v4f*)tx1)[i] = z;
  ((v4f*)tx2)[i] = -xv;
}

// ---------------------------------------------------------------------------
// Kernel 2: COO SpMM with per-edge scatter-atomics.
// One thread per (edge, float4-chunk): 128 chunks/edge covering T*C = 512 feats.
// dst[t, row, c] += scale * val[e] * src[t, col, c]
// ---------------------------------------------------------------------------
__global__ void cheb_spmm_kernel(const int* __restrict__ rows,
                                 const int* __restrict__ cols,
                                 const float* __restrict__ vals,
                                 const float* __restrict__ src,
                                 float* __restrict__ dst,
                                 float scale) {
  const long total = (long)E_EDGES * (T_DIM * C_DIM / 4);  // E * 128
  long idx = (long)blockIdx.x * blockDim.x + threadIdx.x;
  if (idx >= total) return;

  const int e  = (int)(idx >> 7);   // / 128
  const int r  = (int)(idx & 127);
  const int t  = r >> 5;            // 0..3
  const int c4 = r & 31;            // float4 chunk within 128 features

  const int   row = rows[e];
  const int   col = cols[e];
  const float v   = vals[e] * scale;

  const v4f s = ((const v4f*)(src + ((long)t * N_NODES + col) * C_DIM))[c4];
  float* d = dst + ((long)t * N_NODES + row) * C_DIM + c4 * 4;
  atomicAdd(d + 0, v * s.x);
  atomicAdd(d + 1, v * s.y);
  atomicAdd(d + 2, v * s.z);
  atomicAdd(d + 3, v * s.w);
}

// ---------------------------------------------------------------------------
// Kernel 3: out[t, m, o] = sum_k TXk[t, m, :] @ W[t, k, :, o] + bias[t, o]
// via V_WMMA_F32_16X16X4_F32 (wave32, exact f32 path).
//
// Per ISA 7.12.2 (32-bit operands), lane = half*16 + r (half = lane>=16):
//   A (16x4):  a[v] = A[r][2*half + v]           -> one aligned 8B load
//   B (4x16):  b[v] = B[2*half + v][r]           -> two strided 4B loads
//   C/D 16x16: c[v] = C[v + 8*half][r], v=0..7
//
// Block = 256 threads = 8 waves; wave w owns output columns [16w, 16w+16).
// Grid = T * (N/16) blocks; block owns 16 output rows for one t.
// ---------------------------------------------------------------------------
__global__ __launch_bounds__(256)
void cheb_gemm_wmma_kernel(const float* __restrict__ x,
                           const float* __restrict__ tx1,
                           const float* __restrict__ tx2,
                           const float* __restrict__ weight,
                           const float* __restrict__ bias,
                           float* __restrict__ out) {
  const int wave   = threadIdx.x >> 5;
  const int lane   = threadIdx.x & 31;
  const int half   = lane >> 4;       // 0 or 1
  const int lr     = lane & 15;       // A-row / B-col / D-col within tile
  const int t      = blockIdx.x / M_TILES;
  const int m_base = (blockIdx.x % M_TILES) * 16;
  const int o_base = wave * 16;

  // Accumulator seeded with bias (broadcast down rows: col = o_base + lr).
  const float b0 = bias[t * C_DIM + o_base + lr];
  v8f acc;
#pragma unroll
  for (int v = 0; v < 8; ++v) acc[v] = b0;

  const float* srcs[3] = { x, tx1, tx2 };

#pragma unroll
  for (int k = 0; k < K_CHEB; ++k) {
    // A: row (m_base + lr), starting K offset 2*half -> 8B-aligned float2 loads.
    const float* A = srcs[k] + ((long)t * N_NODES + m_base + lr) * C_DIM + 2 * half;
    // B: W[t, k, i + 2*half + {0,1}, o_base + lr], row stride = C_DIM.
    const float* W = weight + (((long)t * K_CHEB + k) * C_DIM + 2 * half) * C_DIM
                   + o_base + lr;

#pragma unroll 4
    for (int i = 0; i < C_DIM; i += 4) {
      v2f a = *(const v2f*)(A + i);
      v2f b;
      b.x = W[(long)i * C_DIM];
      b.y = W[(long)(i + 1) * C_DIM];
      acc = __builtin_amdgcn_wmma_f32_16x16x4_f32(
          /*neg_a=*/false, a, /*neg_b=*/false, b,
          /*c_mod=*/(short)0, acc, /*reuse_a=*/false, /*reuse_b=*/false);
    }
  }

  // D element v lives at row (m_base + v + 8*half), col (o_base + lr).
  float* O = out + ((long)t * N_NODES + m_base + 8 * half) * C_DIM + o_base + lr;
#pragma unroll
  for (int v = 0; v < 8; ++v) O[(long)v * C_DIM] = acc[v];
}

// ---------------------------------------------------------------------------
extern "C" void kernel_launch(void* const* d_in, const int* in_sizes, int n_in,
                              void* d_out, int out_size, void* d_ws, size_t ws_size,
                              hipStream_t stream) {
  const float* x      = (const float*)d_in[0];
  const int*   erow   = (const int*)d_in[1];
  const int*   ecol   = (const int*)d_in[2];
  const float* evals  = (const float*)d_in[3];
  const float* weight = (const float*)d_in[4];
  const float* bias   = (const float*)d_in[5];
  float*       out    = (float*)d_out;

  float* tx1 = (float*)d_ws;
  float* tx2 = tx1 + (size_t)T_DIM * N_NODES * C_DIM;

  const long n4 = (long)T_DIM * N_NODES * C_DIM / 4;       // 6.4M float4
  cheb_init_kernel<<<(unsigned)((n4 + 255) / 256), 256, 0, stream>>>(x, tx1, tx2);

  const long sp = (long)E_EDGES * (T_DIM * C_DIM / 4);     // 102.4M threads
  cheb_spmm_kernel<<<(unsigned)((sp + 255) / 256), 256, 0, stream>>>(
      erow, ecol, evals, x, tx1, 1.0f);
  cheb_spmm_kernel<<<(unsigned)((sp + 255) / 256), 256, 0, stream>>>(
      erow, ecol, evals, tx1, tx2, 2.0f);

  cheb_gemm_wmma_kernel<<<T_DIM * M_TILES, 256, 0, stream>>>(
      x, tx1, tx2, weight, bias, out);
}